// GNNEncoder_12154757448413
// MI455X (gfx1250) — compile-verified
//
#include <hip/hip_runtime.h>
#include <hip/hip_bf16.h>

typedef float v2f __attribute__((ext_vector_type(2)));
typedef float v8f __attribute__((ext_vector_type(8)));

#define HID 64
#define KP  (HID / 2)   // number of k-pairs

// ---------------------------------------------------------------- utilities
__global__ void gnn_zero_f32(float* __restrict__ p, int n) {
    int t = blockIdx.x * blockDim.x + threadIdx.x;
    if (t < n) p[t] = 0.0f;
}

__global__ void gnn_deg_kernel(const long long* __restrict__ dst,
                               float* __restrict__ deg, int nE) {
    int t = blockIdx.x * blockDim.x + threadIdx.x;
    if (t < nE) atomicAdd(&deg[(int)dst[t]], 1.0f);
}

__global__ void gnn_count_kernel(const long long* __restrict__ batch,
                                 float* __restrict__ cnt, int nN) {
    int t = blockIdx.x * blockDim.x + threadIdx.x;
    if (t < nN) atomicAdd(&cnt[(int)batch[t]], 1.0f);
}

// 16 threads per edge, each thread moves 4 consecutive features (float4).
__global__ void gnn_scatter_kernel(const float* __restrict__ h,
                                   const long long* __restrict__ src,
                                   const long long* __restrict__ dst,
                                   float* __restrict__ agg, int nE) {
    long long t = (long long)blockIdx.x * blockDim.x + threadIdx.x;
    int e = (int)(t >> 4);
    if (e >= nE) return;
    int part = ((int)t & 15) * 4;
    int s = (int)src[e];
    int d = (int)dst[e];
    const float4 v = *(const float4*)(h + (size_t)s * HID + part);
    float* o = agg + (size_t)d * HID + part;
    atomicAdd(o + 0, v.x);
    atomicAdd(o + 1, v.y);
    atomicAdd(o + 2, v.z);
    atomicAdd(o + 3, v.w);
}

// Stage W[k][n] (row-major 64x64) into LDS swizzled as
//   sW[p*128 + n*2 + j] = W[2p + j][n]
// so a B fragment {W[kk][n], W[kk+1][n]} is one aligned ds_load_b64.
__device__ __forceinline__ void stage_weights_swizzled(float* sW,
                                                       const float* __restrict__ W) {
    for (int i = threadIdx.x; i < KP * HID; i += 256) {
        int p = i >> 6;          // k-pair
        int n = i & 63;          // output column
        float2 w;
        w.x = W[(2 * p) * HID + n];
        w.y = W[(2 * p + 1) * HID + n];
        *(float2*)&sW[i * 2] = w;
    }
}

// ---------------------------------------------------------------- fused SAGE layer
// out[m][n] = relu( (agg[m]/max(deg[m],1)) @ Wl + bl + h[m] @ Wr )
// One wave32 computes a 16x64 output tile via V_WMMA_F32_16X16X4_F32.
__global__ void __launch_bounds__(256)
gnn_sage_gemm(const float* __restrict__ h, const float* agg,
              const float* __restrict__ deg,
              const float* __restrict__ Wl, const float* __restrict__ bl,
              const float* __restrict__ Wr,
              float* out, int nTiles, int nNodes) {
    __shared__ float sWl[HID * HID];
    __shared__ float sWr[HID * HID];
    stage_weights_swizzled(sWl, Wl);
    stage_weights_swizzled(sWr, Wr);
    __syncthreads();

    int tile = blockIdx.x * 8 + (threadIdx.x >> 5);
    if (tile >= nTiles) return;

    int lane = threadIdx.x & 31;
    int half = lane >> 4;      // K-half for A/B fragments
    int r    = lane & 15;      // row-in-tile (A) / col (B, C/D)

    int rowLd = tile * 16 + r;
    if (rowLd >= nNodes) rowLd = nNodes - 1;   // clamp: keep EXEC all-1s
    const float* hrow = h   + (size_t)rowLd * HID;
    const float* arow = agg + (size_t)rowLd * HID;
    float invd = 1.0f / fmaxf(deg[rowLd], 1.0f);

    v8f acc[4];
#pragma unroll
    for (int j = 0; j < 4; ++j) acc[j] = (v8f)(0.0f);

#pragma unroll
    for (int k0 = 0; k0 < HID; k0 += 4) {
        int kk = k0 + 2 * half;
        int p  = kk >> 1;                      // k-pair index into swizzled LDS
        float2 ah2 = *(const float2*)(hrow + kk);
        float2 aa2 = *(const float2*)(arow + kk);
        v2f aH; aH.x = ah2.x;        aH.y = ah2.y;
        v2f aA; aA.x = aa2.x * invd; aA.y = aa2.y * invd;
#pragma unroll
        for (int j = 0; j < 4; ++j) {
            int n = j * 16 + r;
            v2f bL = *(const v2f*)&sWl[(p * HID + n) * 2];
            v2f bR = *(const v2f*)&sWr[(p * HID + n) * 2];
            acc[j] = __builtin_amdgcn_wmma_f32_16x16x4_f32(
                false, aA, false, bL, (short)0, acc[j], false, false);
            acc[j] = __builtin_amdgcn_wmma_f32_16x16x4_f32(
                false, aH, false, bR, (short)0, acc[j], false, false);
        }
    }

#pragma unroll
    for (int j = 0; j < 4; ++j) {
        int n = j * 16 + r;
        float bias = bl[n];
#pragma unroll
        for (int vr = 0; vr < 8; ++vr) {
            int m = tile * 16 + vr + 8 * half;
            if (m < nNodes) {
                float val = fmaxf(acc[j][vr] + bias, 0.0f);
                out[(size_t)m * HID + n] = val;
            }
        }
    }
}

// ---------------------------------------------------------------- head + pooling
// node_h = relu(h @ W + b) -> d_out; also gsum[batch[m]] += node_h[m]
__global__ void __launch_bounds__(256)
gnn_head_gemm(const float* __restrict__ h, const float* __restrict__ W,
              const float* __restrict__ b, const long long* __restrict__ batch,
              float* __restrict__ out, float* __restrict__ gsum,
              int nTiles, int nNodes) {
    __shared__ float sW[HID * HID];
    stage_weights_swizzled(sW, W);
    __syncthreads();

    int tile = blockIdx.x * 8 + (threadIdx.x >> 5);
    if (tile >= nTiles) return;

    int lane = threadIdx.x & 31;
    int half = lane >> 4;
    int r    = lane & 15;

    int rowLd = tile * 16 + r;
    if (rowLd >= nNodes) rowLd = nNodes - 1;
    const float* hrow = h + (size_t)rowLd * HID;

    v8f acc[4];
#pragma unroll
    for (int j = 0; j < 4; ++j) acc[j] = (v8f)(0.0f);

#pragma unroll
    for (int k0 = 0; k0 < HID; k0 += 4) {
        int kk = k0 + 2 * half;
        int p  = kk >> 1;
        float2 a2 = *(const float2*)(hrow + kk);
        v2f aH; aH.x = a2.x; aH.y = a2.y;
#pragma unroll
        for (int j = 0; j < 4; ++j) {
            int n = j * 16 + r;
            v2f bW = *(const v2f*)&sW[(p * HID + n) * 2];
            acc[j] = __builtin_amdgcn_wmma_f32_16x16x4_f32(
                false, aH, false, bW, (short)0, acc[j], false, false);
        }
    }

#pragma unroll
    for (int j = 0; j < 4; ++j) {
        int n = j * 16 + r;
        float bias = b[n];
#pragma unroll
        for (int vr = 0; vr < 8; ++vr) {
            int m = tile * 16 + vr + 8 * half;
            if (m < nNodes) {
                float val = fmaxf(acc[j][vr] + bias, 0.0f);
                out[(size_t)m * HID + n] = val;
                int g = (int)batch[m];
                atomicAdd(&gsum[g * HID + n], val);
            }
        }
    }
}

__global__ void gnn_pool_kernel(const float* __restrict__ gsum,
                                const float* __restrict__ gcnt,
                                float* __restrict__ outg, int nElem) {
    int t = blockIdx.x * blockDim.x + threadIdx.x;
    if (t < nElem) outg[t] = gsum[t] / fmaxf(gcnt[t >> 6], 1.0f);
}

// ---------------------------------------------------------------- launcher
extern "C" void kernel_launch(void* const* d_in, const int* in_sizes, int n_in,
                              void* d_out, int out_size, void* d_ws, size_t ws_size,
                              hipStream_t stream) {
    const float*     x     = (const float*)d_in[0];
    const long long* edge  = (const long long*)d_in[1];
    const long long* batch = (const long long*)d_in[2];
    const float* Wl0  = (const float*)d_in[3];
    const float* bl0  = (const float*)d_in[4];
    const float* Wr0  = (const float*)d_in[5];
    const float* Wl1  = (const float*)d_in[6];
    const float* bl1  = (const float*)d_in[7];
    const float* Wr1  = (const float*)d_in[8];
    const float* Wlin = (const float*)d_in[9];
    const float* blin = (const float*)d_in[10];

    const int N = in_sizes[0] / HID;          // 100000 nodes
    const int E = in_sizes[1] / 2;            // 1.6M edges
    const int G = (out_size - N * HID) / HID; // 64 graphs

    const long long* src = edge;
    const long long* dst = edge + E;

    char*  ws        = (char*)d_ws;
    size_t nodeBytes = (size_t)N * HID * sizeof(float);
    float* agg  = (float*)ws;                          // reused as h2 (in-place)
    float* h1   = (float*)(ws + nodeBytes);
    float* deg  = (float*)(ws + 2 * nodeBytes);
    float* gsum = deg + N;
    float* gcnt = gsum + G * HID;

    const int nTiles  = (N + 15) / 16;
    const int gemmBlk = (nTiles + 7) / 8;
    const int zeroN   = N * HID;
    const int scatBlk = (int)(((long long)E * 16 + 255) / 256);
    dim3 blk(256);

    // ---- zero scratch (deg | gsum | gcnt contiguous) + layer-0 agg
    gnn_zero_f32<<<(zeroN + 255) / 256, blk, 0, stream>>>(agg, zeroN);
    gnn_zero_f32<<<(N + G * HID + G + 255) / 256, blk, 0, stream>>>(deg, N + G * HID + G);

    // ---- degrees (shared by both layers) + graph counts
    gnn_deg_kernel<<<(E + 255) / 256, blk, 0, stream>>>(dst, deg, E);
    gnn_count_kernel<<<(N + 255) / 256, blk, 0, stream>>>(batch, gcnt, N);

    // ---- layer 0: scatter x, fused GEMM -> h1
    gnn_scatter_kernel<<<scatBlk, blk, 0, stream>>>(x, src, dst, agg, E);
    gnn_sage_gemm<<<gemmBlk, blk, 0, stream>>>(x, agg, deg, Wl0, bl0, Wr0, h1, nTiles, N);

    // ---- layer 1: scatter h1, fused GEMM -> h2 (in-place over agg buffer;
    //      each wave finishes reading its rows of agg before storing them)
    gnn_zero_f32<<<(zeroN + 255) / 256, blk, 0, stream>>>(agg, zeroN);
    gnn_scatter_kernel<<<scatBlk, blk, 0, stream>>>(h1, src, dst, agg, E);
    gnn_sage_gemm<<<gemmBlk, blk, 0, stream>>>(h1, agg, deg, Wl1, bl1, Wr1, agg, nTiles, N);

    // ---- head: node_h -> d_out[0:N*HID], pooled sums -> gsum
    float* node_out = (float*)d_out;
    gnn_head_gemm<<<gemmBlk, blk, 0, stream>>>(agg, Wlin, blin, batch,
                                               node_out, gsum, nTiles, N);

    // ---- mean pool -> d_out[N*HID : N*HID + G*HID]
    gnn_pool_kernel<<<(G * HID + 255) / 256, blk, 0, stream>>>(
        gsum, gcnt, node_out + (size_t)N * HID, G * HID);
}